// AttentionLayer_context_3298534883586
// MI455X (gfx1250) — compile-verified
//
#include <hip/hip_runtime.h>
#include <math.h>

// CDNA5 ext-vector types for WMMA operands
typedef float v2f __attribute__((ext_vector_type(2)));
typedef float v8f __attribute__((ext_vector_type(8)));

#define BATCH   256
#define NREG    1024
#define DDIM    512
#define THREADS 256   // 8 wave32 waves

// Branch-free tanh. gfx1250 has a hardware V_TANH_F32 transcendental op;
// fall back to a 1-exp formulation (still branch-free, no saveexec regions).
__device__ __forceinline__ float fast_tanh(float v) {
#if __has_builtin(__builtin_amdgcn_tanhf)
    return __builtin_amdgcn_tanhf(v);
#else
    const float t = __expf(-2.0f * fabsf(v));           // v_exp_f32
    const float r = (1.0f - t) * __builtin_amdgcn_rcpf(1.0f + t);
    return copysignf(r, v);
#endif
}

// One workgroup per batch element b.
//   Phase 1: e[n] = sum_d tanh(x[b,d] + F[b,n,d]) * vw[d]   via V_WMMA_F32_16X16X4_F32
//   Phase 2: alpha = softmax(e) in LDS
//   Phase 3: context[b,d] = sum_n F[b,n,d] * alpha[n]       coalesced VALU matvec (L2 re-hit)
__global__ __launch_bounds__(THREADS)
void bahdanau_attention_kernel(const float* __restrict__ x,      // [B, D]
                               const float* __restrict__ feats,  // [B, N, D]
                               const float* __restrict__ vw,     // [D]
                               const float* __restrict__ vb,     // [1]
                               float* __restrict__ out)          // [B, D]
{
    __shared__ float s_x[DDIM];     // x row for this batch
    __shared__ float s_vw[DDIM];    // score vector
    __shared__ float s_e[NREG];     // scores -> exp -> alpha
    __shared__ float s_red[THREADS];

    const int b    = blockIdx.x;
    const int tid  = threadIdx.x;
    const int wave = tid >> 5;
    const int lane = tid & 31;

    const float* xb = x + (size_t)b * DDIM;
    const float* fb = feats + (size_t)b * NREG * DDIM;
    const float  bias = vb[0];   // cancels in softmax; kept for fidelity of e

    // ---- Phase 0: stage x[b,:] and v_w in LDS ----
    for (int i = tid; i < DDIM; i += THREADS) { s_x[i] = xb[i]; s_vw[i] = vw[i]; }
    __syncthreads();

    // ---- Phase 1: scores via f32 WMMA 16x16x4 ----
    // A-matrix (16x4 f32) lane layout: lanes 0-15 hold K={0,1}, lanes 16-31 hold K={2,3};
    // row M = lane&15. B-matrix (4x16) mirrors the K split across lane halves; our B
    // columns are all identical (v_w replicated), so every row of C carries one
    // region's score, replicated across columns.
    const int row  = lane & 15;
    const int koff = (lane >> 4) << 1;   // 0 or 2

    #pragma unroll 1
    for (int t = 0; t < 8; ++t) {
        const int n0 = (wave * 8 + t) * 16;               // 64 tiles of 16 regions
        const float* frow = fb + (size_t)(n0 + row) * DDIM;

        v8f c = {0.f, 0.f, 0.f, 0.f, 0.f, 0.f, 0.f, 0.f};

        #pragma unroll 4
        for (int d = 0; d < DDIM; d += 4) {
            // contiguous 8B-aligned pairs: one global b64 + two LDS b64 per step
            v2f fa = *(const v2f*)(frow + d + koff);
            v2f xa = *(const v2f*)(&s_x[d + koff]);
            v2f bm = *(const v2f*)(&s_vw[d + koff]);
            v2f a;
            a.x = fast_tanh(xa.x + fa.x);    // v_tanh_f32 (trans pipe, co-executes w/ WMMA)
            a.y = fast_tanh(xa.y + fa.y);
            // (neg_a, A, neg_b, B, c_mod, C, reuse_a, reuse_b)
            c = __builtin_amdgcn_wmma_f32_16x16x4_f32(false, a, false, bm,
                                                      (short)0, c, false, false);
        }
        // C/D layout: VGPR r, lanes 0-15 -> M=r, N=lane; lanes 16-31 -> M=8+r.
        // Column N=0 lives in lanes 0 and 16.
        if (lane == 0) {
            #pragma unroll
            for (int r = 0; r < 8; ++r) s_e[n0 + r] = c[r] + bias;
        } else if (lane == 16) {
            #pragma unroll
            for (int r = 0; r < 8; ++r) s_e[n0 + 8 + r] = c[r] + bias;
        }
    }
    __syncthreads();

    // ---- Phase 2: softmax over s_e[0..N) ----
    float m = -INFINITY;
    for (int i = tid; i < NREG; i += THREADS) m = fmaxf(m, s_e[i]);
    s_red[tid] = m;
    __syncthreads();
    for (int s = THREADS / 2; s > 0; s >>= 1) {
        if (tid < s) s_red[tid] = fmaxf(s_red[tid], s_red[tid + s]);
        __syncthreads();
    }
    const float gmax = s_red[0];
    __syncthreads();   // everyone has gmax before s_red is reused

    float sum = 0.f;
    for (int i = tid; i < NREG; i += THREADS) {
        float ex = __expf(s_e[i] - gmax);   // v_exp_f32 path
        s_e[i] = ex;
        sum += ex;
    }
    s_red[tid] = sum;
    __syncthreads();
    for (int s = THREADS / 2; s > 0; s >>= 1) {
        if (tid < s) s_red[tid] += s_red[tid + s];
        __syncthreads();
    }
    const float inv = 1.0f / s_red[0];
    for (int i = tid; i < NREG; i += THREADS) s_e[i] *= inv;   // alpha
    __syncthreads();

    // ---- Phase 3: context[b, 2t:2t+2] = sum_n F[b,n,:] * alpha[n] ----
    // Adjacent threads read adjacent float2 -> fully coalesced 2KB per region row;
    // rows were just pulled through L2 in phase 1 (2 MB/batch << 192 MB L2).
    const int d0 = tid * 2;
    const float* p = fb + d0;
    float ax = 0.f, ay = 0.f;
    #pragma unroll 4
    for (int n = 0; n < NREG; ++n) {
        if (n + 16 < NREG)
            __builtin_prefetch(p + (size_t)(n + 16) * DDIM, 0, 1); // global_prefetch_b8
        v2f v = *(const v2f*)(p + (size_t)n * DDIM);
        const float alpha = s_e[n];
        ax = fmaf(v.x, alpha, ax);
        ay = fmaf(v.y, alpha, ay);
    }
    out[(size_t)b * DDIM + d0]     = ax;
    out[(size_t)b * DDIM + d0 + 1] = ay;
}

extern "C" void kernel_launch(void* const* d_in, const int* in_sizes, int n_in,
                              void* d_out, int out_size, void* d_ws, size_t ws_size,
                              hipStream_t stream) {
    const float* x     = (const float*)d_in[0];  // [256, 512]
    const float* feats = (const float*)d_in[1];  // [256, 1024, 512]
    const float* vw    = (const float*)d_in[2];  // [1, 512]
    const float* vb    = (const float*)d_in[3];  // [1]
    float* out = (float*)d_out;                  // [256, 512]

    bahdanau_attention_kernel<<<BATCH, THREADS, 0, stream>>>(x, feats, vw, vb, out);
}